// GCN_15590731285080
// MI455X (gfx1250) — compile-verified
//
#include <hip/hip_runtime.h>
#include <stdint.h>

typedef __attribute__((ext_vector_type(2))) float v2f;
typedef __attribute__((ext_vector_type(8))) float v8f;

#define KDIM 128   // inner dim of every layer

// ---------------------------------------------------------------- utilities

__global__ void zero_kernel(float* __restrict__ p, long n) {
    long i = (long)blockIdx.x * blockDim.x + threadIdx.x;
    if (i < n) p[i] = 0.0f;
}

// deg[d] += 1 for every edge destination (self-loop +1 folded into dinv_kernel)
__global__ void deg_kernel(const long long* __restrict__ ei, float* __restrict__ deg, int E) {
    int e = blockIdx.x * blockDim.x + threadIdx.x;
    if (e < E) {
        int d = (int)ei[(long)E + e];
        atomicAdd(&deg[d], 1.0f);
    }
}

// in-place: dinv[i] = rsqrt(deg[i] + 1)   (the +1 is the self-loop; deg>=1 always)
__global__ void dinv_kernel(float* __restrict__ deg, int n) {
    int i = blockIdx.x * blockDim.x + threadIdx.x;
    if (i < n) deg[i] = rsqrtf(deg[i] + 1.0f);
}

__global__ void norm_kernel(const long long* __restrict__ ei, const float* __restrict__ dinv,
                            float* __restrict__ norm, int E) {
    int e = blockIdx.x * blockDim.x + threadIdx.x;
    if (e < E) {
        int s = (int)ei[e];
        int d = (int)ei[(long)E + e];
        norm[e] = dinv[s] * dinv[d];
    }
}

// ---------------------------------------------------------------- WMMA GEMM
// C[M x Ncols] = A[M x K=128] * B[128 x Ncols], row-major, exact fp32 via
// V_WMMA_F32_16X16X4_F32.
//
// blockIdx.y = 16-wide N tile. Each 256-thread block stages its 128x16 B tile
// (8 KB) in LDS once; each of its 8 waves then copies its per-lane B fragment
// into 64 VGPRs and loops over M tiles, so the hot loop does only A loads +
// 32 chained WMMAs (B never touches memory again).
__global__ void gemm_wmma_f32(const float* __restrict__ A, const float* __restrict__ B,
                              float* __restrict__ C, int M, int ldb, int Ncols, int ldc) {
    __shared__ float Bs[KDIM * 16];                       // 8 KB of the 320 KB/WGP LDS

    const int tid = threadIdx.x;
    const int tN  = blockIdx.y;
    const int colBase = tN * 16;

    // cooperative fill: 2048 elements / 256 threads = 8 each (masked pad cols -> 0)
#pragma unroll
    for (int j = 0; j < 8; ++j) {
        int idx = tid + j * 256;
        int k   = idx >> 4;
        int c   = idx & 15;
        int col = colBase + c;
        Bs[idx] = (col < Ncols) ? B[(size_t)k * ldb + col] : 0.0f;
    }
    __syncthreads();

    const int lane = tid & 31;
    const int wv   = tid >> 5;                            // wave in block, 0..7
    const int half = lane >> 4;                           // A: half 0 holds K{0,1}, half 1 K{2,3}
    const int r    = lane & 15;

    // B fragment for the whole K loop, resident in registers (64 VGPRs)
    v2f bf[KDIM / 4];
#pragma unroll
    for (int kk = 0; kk < KDIM / 4; ++kk) {
        int k0 = kk * 4 + 2 * half;
        bf[kk].x = Bs[k0 * 16 + r];
        bf[kk].y = Bs[(k0 + 1) * 16 + r];
    }

    const int mtiles  = (M + 15) >> 4;
    const int wstride = gridDim.x * 8;

    for (int tM = blockIdx.x * 8 + wv; tM < mtiles; tM += wstride) {
        const float* Arow = A + (size_t)(tM * 16 + r) * KDIM;
        __builtin_prefetch(A + (size_t)((tM + wstride) * 16 + r) * KDIM, 0, 3);

        v8f acc = {0.f, 0.f, 0.f, 0.f, 0.f, 0.f, 0.f, 0.f};
#pragma unroll
        for (int kk = 0; kk < KDIM / 4; ++kk) {
            int k0 = kk * 4 + 2 * half;
            v2f a;
            a.x = Arow[k0];
            a.y = Arow[k0 + 1];
            acc = __builtin_amdgcn_wmma_f32_16x16x4_f32(
                false, a, false, bf[kk], (short)0, acc, false, false);
        }

        // C/D layout: VGPR v holds row M=v (lanes 0-15) and M=v+8 (lanes 16-31)
        float* Crow = C + (size_t)(tM * 16 + half * 8) * ldc + colBase + r;
#pragma unroll
        for (int v = 0; v < 8; ++v) Crow[(size_t)v * ldc] = acc[v];
    }
}

// ---------------------------------------------------------------- aggregation
// out[dst] += h[src] * w   for all E real edges + N implicit self loops.
// F=128: one wave per edge, one float4 per lane (32*4 = 128); both h and out
// are L2-resident (51.2 MB each vs 192 MB L2).
__global__ void agg128(const float* __restrict__ h, const long long* __restrict__ ei,
                       const float* __restrict__ norm, const float* __restrict__ dinv,
                       float* __restrict__ out, int E, int n) {
    int wid  = (int)((blockIdx.x * blockDim.x + threadIdx.x) >> 5);
    int lane = threadIdx.x & 31;
    if (wid >= E + n) return;

    int s, d; float w;
    if (wid < E) {
        s = (int)ei[wid];
        d = (int)ei[(long)E + wid];
        w = norm[wid];
    } else {
        s = d = wid - E;
        float di = dinv[s];
        w = di * di;
    }

    const float4 v = ((const float4*)(h + (size_t)s * 128))[lane];
    float* op = out + (size_t)d * 128 + lane * 4;
    atomicAdd(op + 0, v.x * w);
    atomicAdd(op + 1, v.y * w);
    atomicAdd(op + 2, v.z * w);
    atomicAdd(op + 3, v.w * w);
}

// F=40 (src rows padded to stride 48, dst stride 40): lane covers f=lane and f=lane+32(<40)
__global__ void agg40(const float* __restrict__ h, const long long* __restrict__ ei,
                      const float* __restrict__ norm, const float* __restrict__ dinv,
                      float* __restrict__ out, int E, int n) {
    int wid  = (int)((blockIdx.x * blockDim.x + threadIdx.x) >> 5);
    int lane = threadIdx.x & 31;
    if (wid >= E + n) return;

    int s, d; float w;
    if (wid < E) {
        s = (int)ei[wid];
        d = (int)ei[(long)E + wid];
        w = norm[wid];
    } else {
        s = d = wid - E;
        float di = dinv[s];
        w = di * di;
    }

    const float* hp = h + (size_t)s * 48;
    float*       op = out + (size_t)d * 40;
    atomicAdd(op + lane, hp[lane] * w);
    if (lane < 8) atomicAdd(op + 32 + lane, hp[32 + lane] * w);
}

// ---------------------------------------------------------------- bias (+relu)
__global__ void bias_act(float* __restrict__ x, const float* __restrict__ b,
                         long total, int F, int relu) {
    long i = (long)blockIdx.x * blockDim.x + threadIdx.x;
    if (i < total) {
        float v = x[i] + b[(int)(i % F)];
        x[i] = relu ? fmaxf(v, 0.0f) : v;
    }
}

// ---------------------------------------------------------------- launcher
extern "C" void kernel_launch(void* const* d_in, const int* in_sizes, int n_in,
                              void* d_out, int out_size, void* d_ws, size_t ws_size,
                              hipStream_t stream) {
    const float*     x  = (const float*)d_in[0];
    const long long* ei = (const long long*)d_in[1];
    const float*     W1 = (const float*)d_in[2];
    const float*     b1 = (const float*)d_in[3];
    const float*     W2 = (const float*)d_in[4];
    const float*     b2 = (const float*)d_in[5];
    const float*     W3 = (const float*)d_in[6];
    const float*     b3 = (const float*)d_in[7];
    float*           out = (float*)d_out;

    const int n = in_sizes[0] / 128;   // 100000 nodes
    const int E = in_sizes[1] / 2;     // 1600000 edges

    // workspace layout (floats)
    float* dinv = (float*)d_ws;                 // n
    float* norm = dinv + n;                     // E
    float* gbuf = norm + E;                     // n*128 (gemm out; layer3 uses n*48)
    float* abuf = gbuf + (size_t)n * 128;       // n*128 (aggregate out / next input)

    const int BT = 256;
    auto blocks = [](long work, int bt) { return (unsigned)((work + bt - 1) / bt); };

    // ---- normalization: deg -> dinv -> per-edge norm -------------------
    zero_kernel<<<blocks(n, BT), BT, 0, stream>>>(dinv, n);
    deg_kernel<<<blocks(E, BT), BT, 0, stream>>>(ei, dinv, E);
    dinv_kernel<<<blocks(n, BT), BT, 0, stream>>>(dinv, n);
    norm_kernel<<<blocks(E, BT), BT, 0, stream>>>(ei, dinv, norm, E);

    const long aggThr = (long)(E + n) * 32;           // one wave per edge
    const dim3 g128(196, 8);                          // 6250 M tiles / (196*8 waves) ~ 4 tiles/wave
    const dim3 g40 (196, 3);

    // ---- layer 1: gemm(x,W1) -> gbuf ; aggregate -> abuf ; bias+relu ---
    gemm_wmma_f32<<<g128, BT, 0, stream>>>(x, W1, gbuf, n, 128, 128, 128);
    zero_kernel<<<blocks((long)n * 128, BT), BT, 0, stream>>>(abuf, (long)n * 128);
    agg128<<<blocks(aggThr, BT), BT, 0, stream>>>(gbuf, ei, norm, dinv, abuf, E, n);
    bias_act<<<blocks((long)n * 128, BT), BT, 0, stream>>>(abuf, b1, (long)n * 128, 128, 1);

    // ---- layer 2 -------------------------------------------------------
    gemm_wmma_f32<<<g128, BT, 0, stream>>>(abuf, W2, gbuf, n, 128, 128, 128);
    zero_kernel<<<blocks((long)n * 128, BT), BT, 0, stream>>>(abuf, (long)n * 128);
    agg128<<<blocks(aggThr, BT), BT, 0, stream>>>(gbuf, ei, norm, dinv, abuf, E, n);
    bias_act<<<blocks((long)n * 128, BT), BT, 0, stream>>>(abuf, b2, (long)n * 128, 128, 1);

    // ---- layer 3: Ncols=40 padded to 3 tiles (48), ldc=48 --------------
    gemm_wmma_f32<<<g40, BT, 0, stream>>>(abuf, W3, gbuf, n, 40, 40, 48);
    zero_kernel<<<blocks((long)n * 40, BT), BT, 0, stream>>>(out, (long)n * 40);
    agg40<<<blocks(aggThr, BT), BT, 0, stream>>>(gbuf, ei, norm, dinv, out, E, n);
    bias_act<<<blocks((long)n * 40, BT), BT, 0, stream>>>(out, b3, (long)n * 40, 40, 0);
}